// MagicKernelSharp2021Upsample_12292196401644
// MI455X (gfx1250) — compile-verified
//
#include <hip/hip_runtime.h>
#include <hip/hip_bf16.h>
#include <stdint.h>

// MagicKernelSharp2021 + NN-upsample(2x) + B-spline blur, fused as:
//   pass 1: mid = 0.5*x + 0.5*sep7(x)          (256x256, WMMA f32 vertical pass)
//   pass 2: out = polyphase 3-tap x2 of mid    (512x512, VALU, async LDS stage,
//                                               NT stores)
// mid lives in d_ws (needs 8*64*256*256*4 = 128 MiB) and stays resident in the
// 192 MB L2 between the passes; output stores are non-temporal to not evict it.

typedef __attribute__((ext_vector_type(2))) float v2f;
typedef __attribute__((ext_vector_type(8))) float v8f;

#define NH 256
#define NW 256
#define NH2 512
#define NW2 512

// SHARP = [-1/32, 0, 9/32, 16/32, 9/32, 0, -1/32]
// band(m,k) = SHARP[k-m]; table indexed by (k-m)+16, d in [-15,23] -> idx 1..39
__constant__ float BAND_TAB[40] = {
    0.f, 0.f, 0.f, 0.f, 0.f, 0.f, 0.f, 0.f,
    0.f, 0.f, 0.f, 0.f, 0.f, 0.f, 0.f, 0.f,
    -0.03125f,          // d = 0
    0.f,                // d = 1
    0.28125f,           // d = 2
    0.5f,               // d = 3
    0.28125f,           // d = 4
    0.f,                // d = 5
    -0.03125f,          // d = 6
    0.f, 0.f, 0.f, 0.f, 0.f, 0.f, 0.f, 0.f,
    0.f, 0.f, 0.f, 0.f, 0.f, 0.f, 0.f, 0.f, 0.f};

// ---------------- pass 1: sharpen + blend -> mid ----------------
// one wave per 16x16 mid tile; inter = horizontal sharpen over 22 rows (pad 24)
__global__ __launch_bounds__(256) void k_sharpen(const float* __restrict__ x,
                                                 float* __restrict__ mid) {
  const int lane = threadIdx.x & 31;
  const int wv   = threadIdx.x >> 5;
  const int gt   = blockIdx.x * 8 + wv;              // 131072 tiles total
  const int tc   = gt & 15;
  const int tr   = (gt >> 4) & 15;
  const int pl   = gt >> 8;                          // 0..511 (N*C planes)
  const int r0   = tr << 4, c0 = tc << 4;
  const float* plane  = x   + (size_t)pl * (NH * NW);
  float*       mplane = mid + (size_t)pl * (NH * NW);

  __shared__ float inter[8][24][16];                 // 12 KiB / block
  float (*iw)[16] = inter[wv];

  // horizontal 7-tap: lanes 0..23 own inter rows (rows 22,23 forced zero)
  if (lane < 24) {
    const int gr  = r0 - 3 + lane;
    const bool rok = (lane < 22) && (gr >= 0) && (gr < NH);
    const int cb  = c0 - 4;                          // 16B-aligned column base
    float r[24];
    if (rok && cb >= 0 && (cb + 24) <= NW) {
      const float4* p = (const float4*)(plane + (size_t)gr * NW + cb);
#pragma unroll
      for (int k = 0; k < 6; ++k) {
        float4 v = p[k];
        r[4*k+0] = v.x; r[4*k+1] = v.y; r[4*k+2] = v.z; r[4*k+3] = v.w;
      }
    } else {
#pragma unroll
      for (int k = 0; k < 24; ++k) {
        int gc = cb + k;
        r[k] = (rok && gc >= 0 && gc < NW) ? plane[(size_t)gr * NW + gc] : 0.0f;
      }
    }
#pragma unroll
    for (int n = 0; n < 16; ++n) {
      float s = -0.03125f * (r[n+1] + r[n+7]);
      s = fmaf(0.28125f, r[n+3] + r[n+5], s);
      s = fmaf(0.5f,     r[n+4],          s);
      iw[lane][n] = s;
    }
  }
  __syncthreads();

  // vertical 7-tap as banded GEMM: Mid(16x16) = A(16x24) * Inter(24x16)
  const int half = lane >> 4;   // K-half / M-half selector
  const int m    = lane & 15;   // row of A / col of B,C
  v8f acc = {0.f, 0.f, 0.f, 0.f, 0.f, 0.f, 0.f, 0.f};

#if __has_builtin(__builtin_amdgcn_wmma_f32_16x16x4_f32)
#pragma unroll
  for (int j = 0; j < 6; ++j) {
    const int kk = 4 * j + 2 * half;
    v2f a, b;
    a.x = BAND_TAB[kk - m + 16];       // branchless banded-A generation
    a.y = BAND_TAB[kk + 1 - m + 16];
    b.x = iw[kk    ][m];
    b.y = iw[kk + 1][m];
    acc = __builtin_amdgcn_wmma_f32_16x16x4_f32(false, a, false, b,
                                                (short)0, acc, false, false);
  }
#else
  // VALU fallback (kept so the file always compiles; WMMA is the target path)
#pragma unroll
  for (int v = 0; v < 8; ++v) {
    const int mm = v + 8 * half;
    float s = 0.f;
#pragma unroll
    for (int t = 0; t < 7; ++t) s = fmaf(BAND_TAB[t + 16], iw[mm + t][m], s);
    acc[v] = s;
  }
#endif

  // blend with alpha=0.5 and store mid (regular stores: we WANT L2 residency)
#pragma unroll
  for (int v = 0; v < 8; ++v) {
    const int mm = v + 8 * half;
    const size_t off = (size_t)(r0 + mm) * NW + (c0 + m);
    mplane[off] = 0.5f * (acc[v] + plane[off]);
  }
}

// ---------------- pass 2: polyphase x2 upsample-blur ----------------
// one wave per 16x16 mid tile -> 32x32 output tile
__global__ __launch_bounds__(256) void k_upblur(const float* __restrict__ mid,
                                                float* __restrict__ out) {
  const int lane = threadIdx.x & 31;
  const int wv   = threadIdx.x >> 5;
  const int gt   = blockIdx.x * 8 + wv;
  const int tc   = gt & 15;
  const int tr   = (gt >> 4) & 15;
  const int pl   = gt >> 8;
  const int r0   = tr << 4, c0 = tc << 4;
  const float* mplane = mid + (size_t)pl * (NH * NW);
  float*       oplane = out + (size_t)pl * (size_t)(NH2 * NW2);

  __shared__ float mt[8][18][20];                    // 11.25 KiB / block
  float (*mw)[20] = mt[wv];

  // stage 18x18 mid tile (+1 halo). Interior tiles: CDNA5 async copy to LDS.
  const bool interior = (tr >= 1) && (tr <= 14) && (tc >= 1) && (tc <= 14);
  if (interior) {
    const float* gbase = mplane + (size_t)(r0 - 1) * NW + (c0 - 1);
#pragma unroll
    for (int t = 0; t < 11; ++t) {
      int idx = t * 32 + lane;
      if (idx < 324) {
        int row = idx / 18, col = idx - row * 18;
        // low 32 bits of the generic shared pointer == LDS byte offset
        unsigned ldsoff = (unsigned)(uintptr_t)&mw[row][col];
        uint64_t gaddr  = (uint64_t)(uintptr_t)(gbase + row * NW + col);
        asm volatile("global_load_async_to_lds_b32 %0, %1, off"
                     :: "v"(ldsoff), "v"(gaddr)
                     : "memory");
      }
    }
    asm volatile("s_wait_asynccnt 0x0" ::: "memory");
  } else {
#pragma unroll
    for (int t = 0; t < 11; ++t) {
      int idx = t * 32 + lane;
      if (idx < 324) {
        int row = idx / 18, col = idx - row * 18;
        int gr = r0 - 1 + row, gc = c0 - 1 + col;
        float v = 0.f;
        if (gr >= 0 && gr < NH && gc >= 0 && gc < NW)
          v = mplane[(size_t)gr * NW + gc];
        mw[row][col] = v;
      }
    }
  }
  __syncthreads();

  // lane -> output column 2*c0+lane; j = c0 + (lane>>1), phase = lane&1
  const int   jl = lane >> 1;
  const int   pc = lane & 1;
  const float w0 = pc ? 1.f : 5.f;
  const float w2 = pc ? 5.f : 1.f;

  float a0 = mw[0][jl], a1 = mw[0][jl+1], a2 = mw[0][jl+2];
  float b0 = mw[1][jl], b1 = mw[1][jl+1], b2 = mw[1][jl+2];
  float e0 = mw[2][jl], e1 = mw[2][jl+1], e2 = mw[2][jl+2];

  float* obase = oplane + (size_t)(2 * r0) * NW2 + (2 * c0 + lane);

#pragma unroll
  for (int p = 0; p < 16; ++p) {
    // vertical polyphase: even [5,10,1]/16, odd [1,10,5]/16 (scale folded below)
    float ve0 = fmaf(5.f, a0, fmaf(10.f, b0, e0));
    float ve1 = fmaf(5.f, a1, fmaf(10.f, b1, e1));
    float ve2 = fmaf(5.f, a2, fmaf(10.f, b2, e2));
    float vo0 = fmaf(5.f, e0, fmaf(10.f, b0, a0));
    float vo1 = fmaf(5.f, e1, fmaf(10.f, b1, a1));
    float vo2 = fmaf(5.f, e2, fmaf(10.f, b2, a2));
    float oe = fmaf(w0, ve0, fmaf(10.f, ve1, w2 * ve2)) * (1.0f / 256.0f);
    float oo = fmaf(w0, vo0, fmaf(10.f, vo1, w2 * vo2)) * (1.0f / 256.0f);
    // non-temporal: don't let the 512 MiB output stream evict mid from L2
    __builtin_nontemporal_store(oe, &obase[(size_t)(2 * p)     * NW2]);
    __builtin_nontemporal_store(oo, &obase[(size_t)(2 * p + 1) * NW2]);
    if (p < 15) {
      a0 = b0; a1 = b1; a2 = b2;
      b0 = e0; b1 = e1; b2 = e2;
      e0 = mw[p+3][jl]; e1 = mw[p+3][jl+1]; e2 = mw[p+3][jl+2];
    }
  }
}

extern "C" void kernel_launch(void* const* d_in, const int* in_sizes, int n_in,
                              void* d_out, int out_size, void* d_ws, size_t ws_size,
                              hipStream_t stream) {
  (void)in_sizes; (void)n_in; (void)out_size; (void)ws_size;
  const float* x   = (const float*)d_in[0];
  float*       out = (float*)d_out;
  float*       mid = (float*)d_ws;   // needs 128 MiB scratch

  const int tiles  = 8 * 64 * 16 * 16;   // 131072 mid tiles
  const int blocks = tiles / 8;          // 8 waves (256 thr) per block
  hipLaunchKernelGGL(k_sharpen, dim3(blocks), dim3(256), 0, stream, x, mid);
  hipLaunchKernelGGL(k_upblur,  dim3(blocks), dim3(256), 0, stream, mid, out);
}